// TypeSpecificEncoder_67817533604355
// MI455X (gfx1250) — compile-verified
//
#include <hip/hip_runtime.h>

typedef __attribute__((ext_vector_type(2))) float v2f;
typedef __attribute__((ext_vector_type(8))) float v8f;

#define NUM_TYPES 4
#define IN_DIM    128
#define HID_DIM   256
#define OUT_DIM   128
#define M_TILE    64

#define XS_STRIDE (IN_DIM + 4)   // 132 ≡ 4 (mod 64 banks) -> conflict-free b64 reads
#define HS_STRIDE (HID_DIM + 4)  // 260 ≡ 4 (mod 64 banks)

// ---------------- bucketing kernels ----------------

__global__ void k_init(int* cnt) {
  if (threadIdx.x < NUM_TYPES) cnt[threadIdx.x] = 0;
}

__global__ void k_hist(const int* __restrict__ type, int* cnt, int N) {
  int n = blockIdx.x * blockDim.x + threadIdx.x;
  if (n < N) atomicAdd(&cnt[type[n]], 1);
}

__global__ void k_scan(const int* __restrict__ cnt, int* basev, int* running) {
  if (threadIdx.x == 0) {
    int acc = 0;
    for (int t = 0; t < NUM_TYPES; ++t) { basev[t] = acc; running[t] = acc; acc += cnt[t]; }
    basev[NUM_TYPES] = acc;
  }
}

__global__ void k_scatter(const int* __restrict__ type, int* running, int* gidx, int N) {
  int n = blockIdx.x * blockDim.x + threadIdx.x;
  if (n < N) { int p = atomicAdd(&running[type[n]], 1); gidx[p] = n; }
}

// W1[t][in][hid] -> W1T[t][hid][in];  W2[t][hid][out] -> W2T[t][out][hid]
__global__ void k_transpose(const float* __restrict__ W1, const float* __restrict__ W2,
                            float* __restrict__ W1T, float* __restrict__ W2T) {
  int i = blockIdx.x * blockDim.x + threadIdx.x;
  const int tot = NUM_TYPES * IN_DIM * HID_DIM; // same element count for both
  if (i < tot) {
    int t = i / (IN_DIM * HID_DIM);
    int r = (i / HID_DIM) % IN_DIM;   // in
    int c = i % HID_DIM;              // hid
    W1T[((size_t)t * HID_DIM + c) * IN_DIM + r] = W1[i];
    int t2 = i / (HID_DIM * OUT_DIM);
    int r2 = (i / OUT_DIM) % HID_DIM; // hid
    int c2 = i % OUT_DIM;             // out
    W2T[((size_t)t2 * OUT_DIM + c2) * HID_DIM + r2] = W2[i];
  }
}

// ---------------- grouped-GEMM MLP kernel, M-blocked x4 ----------------

__global__ __launch_bounds__(256) void k_mlp(
    const float* __restrict__ X, const float* __restrict__ b1, const float* __restrict__ b2,
    const float* __restrict__ W1T, const float* __restrict__ W2T,
    const int* __restrict__ basev, const int* __restrict__ gidx,
    float* __restrict__ out) {
  __shared__ float Xs[M_TILE][XS_STRIDE];
  __shared__ float Hs[M_TILE][HS_STRIDE];
  __shared__ int   rows[M_TILE];

  // map block -> (expert type, local 64-row tile within that type's segment)
  int tile = blockIdx.x;
  int ty = -1, ltile = 0, acc = 0, segStart = 0, segCount = 0;
  for (int q = 0; q < NUM_TYPES; ++q) {
    int c  = basev[q + 1] - basev[q];
    int nt = (c + (M_TILE - 1)) >> 6;
    if (tile < acc + nt) { ty = q; ltile = tile - acc; segStart = basev[q]; segCount = c; break; }
    acc += nt;
  }
  if (ty < 0) return;

  int row0 = ltile << 6;
  int tid  = threadIdx.x;

  if (tid < M_TILE) {
    int g = row0 + tid;
    rows[tid] = (g < segCount) ? gidx[segStart + g] : -1;
  }
  __syncthreads();

  // stage 64x128 gathered X tile into LDS (float4 loads)
  for (int i = tid; i < M_TILE * (IN_DIM / 4); i += 256) {
    int r  = i >> 5;      // row 0..63
    int c4 = i & 31;      // float4 column 0..31
    float4 v = make_float4(0.f, 0.f, 0.f, 0.f);
    int node = rows[r];
    if (node >= 0) v = ((const float4*)(X + (size_t)node * IN_DIM))[c4];
    *(float4*)&Xs[r][c4 * 4] = v;
  }
  __syncthreads();

  int wave = tid >> 5;               // 0..7
  int lane = tid & 31;
  int n    = lane & 15;              // A-row / B-col / D-col for this lane
  int kh   = (lane >> 4) << 1;       // K-pair base: 0 or 2
  int mh   = (lane >> 4) << 3;       // D-row base: 0 or 8

  // ---- layer 1: Hs = relu(Xs @ W1[ty] + b1[ty]); 2 N-tiles per wave, 4 M-subtiles ----
  for (int nt = wave; nt < HID_DIM / 16; nt += 8) {
    v8f c0 = {}, c1 = {}, c2 = {}, c3 = {};
    const float* wcol = W1T + ((size_t)ty * HID_DIM + nt * 16 + n) * IN_DIM;
#pragma unroll 4
    for (int k0 = 0; k0 < IN_DIM; k0 += 4) {
      v2f b  = *(const v2f*)&wcol[k0 + kh];           // one B fetch ...
      v2f a0 = *(const v2f*)&Xs[ 0 + n][k0 + kh];     // ... feeds 4 WMMAs
      v2f a1 = *(const v2f*)&Xs[16 + n][k0 + kh];
      v2f a2 = *(const v2f*)&Xs[32 + n][k0 + kh];
      v2f a3 = *(const v2f*)&Xs[48 + n][k0 + kh];
      c0 = __builtin_amdgcn_wmma_f32_16x16x4_f32(false, a0, false, b, (short)0, c0, false, false);
      c1 = __builtin_amdgcn_wmma_f32_16x16x4_f32(false, a1, false, b, (short)0, c1, false, false);
      c2 = __builtin_amdgcn_wmma_f32_16x16x4_f32(false, a2, false, b, (short)0, c2, false, false);
      c3 = __builtin_amdgcn_wmma_f32_16x16x4_f32(false, a3, false, b, (short)0, c3, false, false);
    }
    float bias = b1[ty * HID_DIM + nt * 16 + n];
    int col = nt * 16 + n;
#pragma unroll
    for (int r = 0; r < 8; ++r) {
      float v0 = c0[r] + bias, v1 = c1[r] + bias, v2 = c2[r] + bias, v3 = c3[r] + bias;
      Hs[ 0 + r + mh][col] = v0 > 0.f ? v0 : 0.f;
      Hs[16 + r + mh][col] = v1 > 0.f ? v1 : 0.f;
      Hs[32 + r + mh][col] = v2 > 0.f ? v2 : 0.f;
      Hs[48 + r + mh][col] = v3 > 0.f ? v3 : 0.f;
    }
  }
  __syncthreads();

  // ---- layer 2: out = Hs @ W2[ty] + b2[ty]; 1 N-tile per wave, 4 M-subtiles ----
  {
    int ot = wave;
    v8f c0 = {}, c1 = {}, c2 = {}, c3 = {};
    const float* wcol = W2T + ((size_t)ty * OUT_DIM + ot * 16 + n) * HID_DIM;
#pragma unroll 4
    for (int k0 = 0; k0 < HID_DIM; k0 += 4) {
      v2f b  = *(const v2f*)&wcol[k0 + kh];
      v2f a0 = *(const v2f*)&Hs[ 0 + n][k0 + kh];
      v2f a1 = *(const v2f*)&Hs[16 + n][k0 + kh];
      v2f a2 = *(const v2f*)&Hs[32 + n][k0 + kh];
      v2f a3 = *(const v2f*)&Hs[48 + n][k0 + kh];
      c0 = __builtin_amdgcn_wmma_f32_16x16x4_f32(false, a0, false, b, (short)0, c0, false, false);
      c1 = __builtin_amdgcn_wmma_f32_16x16x4_f32(false, a1, false, b, (short)0, c1, false, false);
      c2 = __builtin_amdgcn_wmma_f32_16x16x4_f32(false, a2, false, b, (short)0, c2, false, false);
      c3 = __builtin_amdgcn_wmma_f32_16x16x4_f32(false, a3, false, b, (short)0, c3, false, false);
    }
    float bias = b2[ty * OUT_DIM + ot * 16 + n];
    int col = ot * 16 + n;
#pragma unroll
    for (int r = 0; r < 8; ++r) {
      int n0 = rows[ 0 + r + mh];
      int n1 = rows[16 + r + mh];
      int n2 = rows[32 + r + mh];
      int n3 = rows[48 + r + mh];
      if (n0 >= 0) out[(size_t)n0 * OUT_DIM + col] = c0[r] + bias;
      if (n1 >= 0) out[(size_t)n1 * OUT_DIM + col] = c1[r] + bias;
      if (n2 >= 0) out[(size_t)n2 * OUT_DIM + col] = c2[r] + bias;
      if (n3 >= 0) out[(size_t)n3 * OUT_DIM + col] = c3[r] + bias;
    }
  }
}

// ---------------- host launcher ----------------

extern "C" void kernel_launch(void* const* d_in, const int* in_sizes, int n_in,
                              void* d_out, int out_size, void* d_ws, size_t ws_size,
                              hipStream_t stream) {
  const float* X    = (const float*)d_in[0];
  const int*   type = (const int*)  d_in[1];
  const float* W1   = (const float*)d_in[2];
  const float* b1   = (const float*)d_in[3];
  const float* W2   = (const float*)d_in[4];
  const float* b2   = (const float*)d_in[5];
  float* out = (float*)d_out;

  int N = in_sizes[0] / IN_DIM;

  // workspace carve-up (d_ws is 256B aligned)
  int* cnt     = (int*)d_ws;          // 8 ints
  int* basev   = cnt + 8;             // 8 ints (uses 5)
  int* running = basev + 8;           // 8 ints (uses 4)
  int* gidx    = running + 8;         // N ints
  int  gpad    = (N + 63) & ~63;
  float* W1T   = (float*)(gidx + gpad);                      // 4*256*128 floats
  float* W2T   = W1T + (size_t)NUM_TYPES * HID_DIM * IN_DIM; // 4*128*256 floats

  int nblk = (N + 255) / 256;
  k_init<<<1, 64, 0, stream>>>(cnt);
  k_hist<<<nblk, 256, 0, stream>>>(type, cnt, N);
  k_scan<<<1, 32, 0, stream>>>(cnt, basev, running);
  k_scatter<<<nblk, 256, 0, stream>>>(type, running, gidx, N);

  int wtot = NUM_TYPES * IN_DIM * HID_DIM;
  k_transpose<<<(wtot + 255) / 256, 256, 0, stream>>>(W1, W2, W1T, W2T);

  int maxTiles = (N + M_TILE - 1) / M_TILE + NUM_TYPES; // per-type tile-count upper bound
  k_mlp<<<maxTiles, 256, 0, stream>>>(X, b1, b2, W1T, W2T, basev, gidx, out);
}